// TokenSparse_45131516346830
// MI455X (gfx1250) — compile-verified
//
#include <hip/hip_runtime.h>
#include <hip/hip_bf16.h>
#include <math.h>

#define B_ 16
#define L_ 4096
#define C_ 512
#define K_ 2458          // ceil(4096 * 0.6)
#define T_ 1024          // threads in sort block
#define TAILN_ 1638      // L_ - K_
#define TAILP_ 1640      // tail padded to multiple of 8 (pad: idx=0, w=0)

typedef __attribute__((ext_vector_type(2))) float v2f;
typedef __attribute__((ext_vector_type(8))) float v8f;

// ---- helpers: order-preserving float<->uint mapping -------------------------
__device__ __forceinline__ unsigned int f2ord(float f) {
    unsigned int u = __float_as_uint(f);
    return (u & 0x80000000u) ? ~u : (u | 0x80000000u);
}
__device__ __forceinline__ float ord2f(unsigned int o) {
    unsigned int u = (o & 0x80000000u) ? (o ^ 0x80000000u) : ~o;
    return __uint_as_float(u);
}

// ---- Kernel 1: per-batch score build + bitonic sort + outputs ---------------
__global__ __launch_bounds__(T_) void tsp_sort_kernel(
    const float* __restrict__ ax, const float* __restrict__ ay,
    int* __restrict__ order, int* __restrict__ tailidx,
    float* __restrict__ wtail, float* __restrict__ mask_out)
{
    __shared__ unsigned long long keys[L_];   // 32 KB
    __shared__ float red[T_];                 // 4 KB
    __shared__ float s_alpha, s_beta, s_m;

    const int b   = blockIdx.x;
    const int tid = threadIdx.x;
    const float* x = ax + b * L_;
    const float* y = ay + b * L_;

    // means of x and y
    float sx = 0.f, sy = 0.f;
    for (int i = tid; i < L_; i += T_) { sx += x[i]; sy += y[i]; }
    red[tid] = sx; __syncthreads();
    for (int s = T_ / 2; s > 0; s >>= 1) { if (tid < s) red[tid] += red[tid + s]; __syncthreads(); }
    float sumx = red[0]; __syncthreads();
    red[tid] = sy; __syncthreads();
    for (int s = T_ / 2; s > 0; s >>= 1) { if (tid < s) red[tid] += red[tid + s]; __syncthreads(); }
    float sumy = red[0];
    if (tid == 0) {
        float mx = sumx * (1.0f / (float)L_);
        float my = sumy * (1.0f / (float)L_);
        float coverage = my / (mx + 1e-6f);
        float a = 1.0f / (1.0f + expf(-(1.0f - coverage)));  // sigmoid(1 - coverage)
        s_alpha = a; s_beta = 1.0f - a;
    }
    __syncthreads();
    const float alpha = s_alpha, beta = s_beta;

    // pack keys: (~ordered(score)) << 32 | idx  -> ascending sort == stable
    // descending argsort of score (ties broken by smaller idx first)
    for (int i = tid; i < L_; i += T_) {
        float sc = alpha * x[i] + beta * y[i];
        unsigned int inv = ~f2ord(sc);
        keys[i] = ((unsigned long long)inv << 32) | (unsigned int)i;
    }
    __syncthreads();

    // bitonic sort ascending (78 compare-exchange sweeps)
    for (int k = 2; k <= L_; k <<= 1) {
        for (int j = k >> 1; j > 0; j >>= 1) {
            for (int i = tid; i < L_; i += T_) {
                int l = i ^ j;
                if (l > i) {
                    unsigned long long a0 = keys[i], b0 = keys[l];
                    bool asc = ((i & k) == 0);
                    if ((a0 > b0) == asc) { keys[i] = b0; keys[l] = a0; }
                }
            }
            __syncthreads();
        }
    }

    // softmax over tail ranks [K_, L_): max is score at rank K_ (sorted desc)
    if (tid == 0) s_m = ord2f(~(unsigned int)(keys[K_] >> 32));
    __syncthreads();
    const float m = s_m;
    float es = 0.f;
    for (int r = K_ + tid; r < L_; r += T_)
        es += expf(ord2f(~(unsigned int)(keys[r] >> 32)) - m);
    red[tid] = es; __syncthreads();
    for (int s = T_ / 2; s > 0; s >>= 1) { if (tid < s) red[tid] += red[tid + s]; __syncthreads(); }
    const float denom = red[0];

    // write: sorted order + 0/1 mask scatter
    for (int r = tid; r < L_; r += T_) {
        unsigned long long kk = keys[r];
        int idx = (int)(kk & 0xFFFFFFFFu);
        order[b * L_ + r] = idx;
        mask_out[b * L_ + idx] = (r < K_) ? 1.0f : 0.0f;
    }
    // write: padded tail indices + softmax weights (sorted order)
    for (int t = tid; t < TAILP_; t += T_) {
        int r = K_ + t;
        int idx = 0;
        float wv = 0.0f;
        if (r < L_) {
            unsigned long long kk = keys[r];
            idx = (int)(kk & 0xFFFFFFFFu);
            wv  = expf(ord2f(~(unsigned int)(kk >> 32)) - m) / denom;
        }
        tailidx[b * TAILP_ + t] = idx;
        wtail[b * TAILP_ + t]   = wv;
    }
}

// ---- Kernel 2: coalesced row gather for select_tokens -----------------------
// one 128-thread block per kept row; 128 * float4 = 512 floats = one row
__global__ __launch_bounds__(128) void tsp_gather_kernel(
    const float* __restrict__ tokens, const int* __restrict__ order,
    float* __restrict__ sel)
{
    const int blk = blockIdx.x;
    const int b = blk / K_;
    const int r = blk - b * K_;
    const int idx = order[b * L_ + r];
    const float4* __restrict__ src =
        (const float4*)(tokens + ((size_t)b * L_ + (size_t)idx) * C_);
    float4* __restrict__ dst = (float4*)(sel + ((size_t)b * K_ + (size_t)r) * C_);
    dst[threadIdx.x] = src[threadIdx.x];
}

// ---- Kernel 3: extra_token = wtail (1xTAIL) x gathered tail rows (TAILxC)
// via WMMA f32 16x16x4. One wave per (batch, 16-channel tile).
// A(16x4) = weight chunk replicated down M; B(4x16) = 4 gathered token rows
// x 16 channels. Row 0 of D = result.
// A layout: lanes 0-15 = M, VGPR0 holds K0 (low half) / K2 (high half), VGPR1 K1/K3.
// B layout: VGPR v, lanes 0-15 = K=v (N=lane), lanes 16-31 = K=v+2 (N=lane-16).
__global__ __launch_bounds__(32) void tsp_extra_kernel(
    const float* __restrict__ tokens, const int* __restrict__ tailidx,
    const float* __restrict__ wtail, float* __restrict__ extra)
{
    const int b    = blockIdx.y;
    const int c0   = blockIdx.x * 16;
    const int lane = threadIdx.x;
    const int n    = lane & 15;
    const int klo  = (lane >> 4) * 2;      // 0 for lanes 0-15, 2 for lanes 16-31
    const int c    = c0 + n;

    const int*   __restrict__ ti = tailidx + b * TAILP_;
    const float* __restrict__ wt = wtail + b * TAILP_;
    const float* __restrict__ tb = tokens + (size_t)b * L_ * C_;

    v8f acc0 = {}; v8f acc1 = {};
    for (int r0 = 0; r0 < TAILP_; r0 += 8) {
        const int ka = r0 + klo;
        const int kb = r0 + 4 + klo;
        const int ia0 = ti[ka], ia1 = ti[ka + 1];
        const int ib0 = ti[kb], ib1 = ti[kb + 1];
        v2f a0; a0.x = wt[ka];  a0.y = wt[ka + 1];
        v2f b0; b0.x = tb[(size_t)ia0 * C_ + c];
                b0.y = tb[(size_t)ia1 * C_ + c];
        v2f a1; a1.x = wt[kb];  a1.y = wt[kb + 1];
        v2f b1; b1.x = tb[(size_t)ib0 * C_ + c];
                b1.y = tb[(size_t)ib1 * C_ + c];
        acc0 = __builtin_amdgcn_wmma_f32_16x16x4_f32(
                   false, a0, false, b0, (short)0, acc0, false, false);
        acc1 = __builtin_amdgcn_wmma_f32_16x16x4_f32(
                   false, a1, false, b1, (short)0, acc1, false, false);
    }
    // D[0][n] lives in acc[0] of lanes 0-15 (and duplicated in lanes 16-31)
    if (lane < 16) extra[b * C_ + c] = acc0[0] + acc1[0];
}

// ---- launcher ---------------------------------------------------------------
extern "C" void kernel_launch(void* const* d_in, const int* in_sizes, int n_in,
                              void* d_out, int out_size, void* d_ws, size_t ws_size,
                              hipStream_t stream) {
    (void)in_sizes; (void)n_in; (void)out_size; (void)ws_size;
    const float* tokens = (const float*)d_in[0];
    const float* ax     = (const float*)d_in[1];
    const float* ay     = (const float*)d_in[2];

    float* out   = (float*)d_out;
    float* sel   = out;                                  // B*K*C
    float* extra = out + (size_t)B_ * K_ * C_;           // B*C
    float* mask  = extra + (size_t)B_ * C_;              // B*L

    int*   order   = (int*)d_ws;                                 // B*L ints
    int*   tailidx = order + (size_t)B_ * L_;                    // B*TAILP_ ints
    float* wtail   = (float*)(tailidx + (size_t)B_ * TAILP_);    // B*TAILP_ floats

    tsp_sort_kernel<<<B_, T_, 0, stream>>>(ax, ay, order, tailidx, wtail, mask);
    tsp_gather_kernel<<<B_ * K_, 128, 0, stream>>>(tokens, order, sel);
    tsp_extra_kernel<<<dim3(C_ / 16, B_), 32, 0, stream>>>(tokens, tailidx, wtail, extra);
}